// SNet_23235773071493
// MI455X (gfx1250) — compile-verified
//
#include <hip/hip_runtime.h>
#include <hip/hip_bf16.h>

// MI455X / gfx1250, wave32. fp32 GEMMs via V_WMMA_F32_16X16X4_F32.
// K-innermost LDS panels (stride 20 floats: 16B-aligned rows, bank-spread) so
// WMMA fragments are single ds_load_b64 ops. Weight panels staged with
// GLOBAL_LOAD_ASYNC_TO_LDS_B128 (ASYNCcnt) when the builtin exists.

#define NMAT 2048
#define BT   128          // block tile (M and N)
#define KT   16           // K panel per stage
#define LSTR 20           // LDS row stride in floats (16 data + 4 pad)
#define NEG_INF (-__builtin_inff())

typedef __attribute__((ext_vector_type(2))) float v2f;
typedef __attribute__((ext_vector_type(8))) float v8f;
typedef __attribute__((ext_vector_type(4))) int   v4i;

#if __has_builtin(__builtin_amdgcn_global_load_async_to_lds_b128)
#define HAVE_ASYNC_LDS 1
#else
#define HAVE_ASYNC_LDS 0
#endif

__device__ __forceinline__ void async_wait0() {
#if HAVE_ASYNC_LDS
#if __has_builtin(__builtin_amdgcn_s_wait_asynccnt)
  __builtin_amdgcn_s_wait_asynccnt(0);
#else
  asm volatile("s_wait_asynccnt 0" ::: "memory");
#endif
#endif
}

// Stage 16B from global into LDS (pure copy path for weight panels).
// Builtin param types (from hipcc diagnostic): (v4i AS1*, v4i AS3*, imm, imm).
__device__ __forceinline__ void stage16(const float* __restrict__ g, float* l) {
#if HAVE_ASYNC_LDS
  __builtin_amdgcn_global_load_async_to_lds_b128(
      (__attribute__((address_space(1))) v4i*)g,
      (__attribute__((address_space(3))) v4i*)l, 0, 0);
#else
  *(float4*)l = *(const float4*)g;
#endif
}

// ---------------- BN stats ----------------

// Per-column stats of X[2048,2048] (stats over rows). scale/shift per column.
__global__ __launch_bounds__(256) void colstats_kernel(
    const float* __restrict__ X, const float* __restrict__ gamma,
    const float* __restrict__ beta, float* __restrict__ scale,
    float* __restrict__ shift) {
  const int c = blockIdx.x * 256 + threadIdx.x;
  float s = 0.f, q = 0.f;
  for (int r = 0; r < NMAT; ++r) {
    float v = X[(size_t)r * NMAT + c];
    s += v; q += v * v;
  }
  const float inv = 1.0f / NMAT;
  float mean = s * inv;
  float var  = q * inv - mean * mean;
  float sc   = gamma[c] * rsqrtf(var + 1e-5f);
  scale[c] = sc;
  shift[c] = beta[c] - mean * sc;
}

// Per-row stats of Y[2048,2048] (stats over columns). One block per row.
__global__ __launch_bounds__(256) void rowstats_kernel(
    const float* __restrict__ Y, const float* __restrict__ gamma,
    const float* __restrict__ beta, float* __restrict__ scale,
    float* __restrict__ shift) {
  __shared__ float ssum[256];
  __shared__ float ssq[256];
  const int row = blockIdx.x;
  const int tid = threadIdx.x;
  float s = 0.f, q = 0.f;
  for (int c = tid; c < NMAT; c += 256) {
    float v = Y[(size_t)row * NMAT + c];
    s += v; q += v * v;
  }
  ssum[tid] = s; ssq[tid] = q;
  __syncthreads();
  for (int st = 128; st > 0; st >>= 1) {
    if (tid < st) { ssum[tid] += ssum[tid + st]; ssq[tid] += ssq[tid + st]; }
    __syncthreads();
  }
  if (tid == 0) {
    const float inv = 1.0f / NMAT;
    float mean = ssum[0] * inv;
    float var  = ssq[0] * inv - mean * mean;
    float sc   = gamma[row] * rsqrtf(var + 1e-5f);
    scale[row] = sc;
    shift[row] = beta[row] - mean * sc;
  }
}

// ---------------- WMMA GEMMs ----------------
// 8 waves as 2(M) x 4(N); each wave owns 64x32 = 4x2 tiles of
// v_wmma_f32_16x16x4_f32. Panels are K-innermost: As[row][k], Bs[col][k],
// row stride LSTR floats. Fragment {K,K+1} = one ds_load_b64.

// GEMM1: Y[k,n] = sum_m W[k,m] * relu(scale[n]*X[m,n]+shift[n]) + bias[k]
__global__ __launch_bounds__(256) void gemm1_kernel(
    const float* __restrict__ W, const float* __restrict__ X,
    const float* __restrict__ scale, const float* __restrict__ shift,
    const float* __restrict__ bias, float* __restrict__ Y) {
  __shared__ float As[BT * LSTR];   // A[row=M][k]
  __shared__ float Bs[BT * LSTR];   // B[col=N][k]
  const int Mblk = blockIdx.y * BT;
  const int Nblk = blockIdx.x * BT;
  const int tid  = threadIdx.x;
  const int lane = tid & 31;
  const int wid  = tid >> 5;
  const int l15  = lane & 15;
  const int hi   = (lane >> 4) & 1;
  const int wm0  = (wid >> 2) * 64;
  const int wn0  = (wid & 3) * 32;

  v8f acc[4][2];
#pragma unroll
  for (int mt = 0; mt < 4; ++mt)
#pragma unroll
    for (int nt = 0; nt < 2; ++nt)
#pragma unroll
      for (int r = 0; r < 8; ++r) acc[mt][nt][r] = 0.0f;

  for (int kb = 0; kb < NMAT / KT; ++kb) {
    __syncthreads();
    // A = W (row-major [M][K]): pure 16B tile copy -> async to LDS.
#pragma unroll
    for (int it = 0; it < 2; ++it) {
      int idx = tid + it * 256;           // 0..511
      int row = idx >> 2;                 // 0..127
      int c4  = idx & 3;                  // 16B chunk within row
      stage16(W + (size_t)(Mblk + row) * NMAT + kb * KT + c4 * 4,
              &As[row * LSTR + c4 * 4]);
    }
    // B = relu(scale*X+shift) transposed to [col][k]: VGPR path.
#pragma unroll
    for (int it = 0; it < 2; ++it) {
      int idx  = tid + it * 256;          // 0..511
      int col  = idx & 127;               // local col
      int kg   = idx >> 7;                // 0..3 -> k base kg*4
      int gcol = Nblk + col;
      const float sc = scale[gcol];
      const float sh = shift[gcol];
      float4 v;
      v.x = X[(size_t)(kb * KT + kg * 4 + 0) * NMAT + gcol];
      v.y = X[(size_t)(kb * KT + kg * 4 + 1) * NMAT + gcol];
      v.z = X[(size_t)(kb * KT + kg * 4 + 2) * NMAT + gcol];
      v.w = X[(size_t)(kb * KT + kg * 4 + 3) * NMAT + gcol];
      v.x = fmaxf(v.x * sc + sh, 0.f);
      v.y = fmaxf(v.y * sc + sh, 0.f);
      v.z = fmaxf(v.z * sc + sh, 0.f);
      v.w = fmaxf(v.w * sc + sh, 0.f);
      *(float4*)&Bs[col * LSTR + kg * 4] = v;
    }
    async_wait0();
    __syncthreads();

#pragma unroll
    for (int q = 0; q < 4; ++q) {
      const int kbase = q * 4 + hi * 2;
      v2f a[4], b[2];
#pragma unroll
      for (int mt = 0; mt < 4; ++mt)
        a[mt] = *(const v2f*)&As[(wm0 + mt * 16 + l15) * LSTR + kbase];
#pragma unroll
      for (int nt = 0; nt < 2; ++nt)
        b[nt] = *(const v2f*)&Bs[(wn0 + nt * 16 + l15) * LSTR + kbase];
#pragma unroll
      for (int mt = 0; mt < 4; ++mt)
#pragma unroll
        for (int nt = 0; nt < 2; ++nt)
          acc[mt][nt] = __builtin_amdgcn_wmma_f32_16x16x4_f32(
              false, a[mt], false, b[nt], (short)0, acc[mt][nt], false, false);
    }
  }

  // Epilogue: bias per output ROW
#pragma unroll
  for (int mt = 0; mt < 4; ++mt)
#pragma unroll
    for (int nt = 0; nt < 2; ++nt) {
      int rg = Mblk + wm0 + mt * 16 + hi * 8;
      int cg = Nblk + wn0 + nt * 16 + l15;
#pragma unroll
      for (int r = 0; r < 8; ++r)
        Y[(size_t)(rg + r) * NMAT + cg] = acc[mt][nt][r] + bias[rg + r];
    }
}

// GEMM2: Out[i,c] = sum_n relu(scale[i]*Ysrc[i,n]+shift[i]) * W2[c,n] + bias[c]
__global__ __launch_bounds__(256) void gemm2_kernel(
    const float* __restrict__ Ysrc, const float* __restrict__ scale,
    const float* __restrict__ shift, const float* __restrict__ W2,
    const float* __restrict__ bias, float* __restrict__ Out) {
  __shared__ float As[BT * LSTR];   // A[row=i][k=n]
  __shared__ float Bs[BT * LSTR];   // B[col=c][k=n]
  const int Mblk = blockIdx.y * BT;
  const int Nblk = blockIdx.x * BT;
  const int tid  = threadIdx.x;
  const int lane = tid & 31;
  const int wid  = tid >> 5;
  const int l15  = lane & 15;
  const int hi   = (lane >> 4) & 1;
  const int wm0  = (wid >> 2) * 64;
  const int wn0  = (wid & 3) * 32;

  v8f acc[4][2];
#pragma unroll
  for (int mt = 0; mt < 4; ++mt)
#pragma unroll
    for (int nt = 0; nt < 2; ++nt)
#pragma unroll
      for (int r = 0; r < 8; ++r) acc[mt][nt][r] = 0.0f;

  for (int kb = 0; kb < NMAT / KT; ++kb) {
    __syncthreads();
    // B = W2 (row-major [c][n], K-contiguous): pure 16B tile copy -> async.
#pragma unroll
    for (int it = 0; it < 2; ++it) {
      int idx = tid + it * 256;
      int col = idx >> 2;                 // 0..127 (local c)
      int c4  = idx & 3;
      stage16(W2 + (size_t)(Nblk + col) * NMAT + kb * KT + c4 * 4,
              &Bs[col * LSTR + c4 * 4]);
    }
    // A = relu(scale_row*Y+shift_row), K-contiguous: VGPR path, float4 copy.
#pragma unroll
    for (int it = 0; it < 2; ++it) {
      int idx = tid + it * 256;
      int row = idx >> 2;                 // 0..127 (local i)
      int c4  = idx & 3;
      const float sr = scale[Mblk + row];
      const float tr = shift[Mblk + row];
      float4 v = *(const float4*)(Ysrc + (size_t)(Mblk + row) * NMAT + kb * KT + c4 * 4);
      v.x = fmaxf(v.x * sr + tr, 0.f);
      v.y = fmaxf(v.y * sr + tr, 0.f);
      v.z = fmaxf(v.z * sr + tr, 0.f);
      v.w = fmaxf(v.w * sr + tr, 0.f);
      *(float4*)&As[row * LSTR + c4 * 4] = v;
    }
    async_wait0();
    __syncthreads();

#pragma unroll
    for (int q = 0; q < 4; ++q) {
      const int kbase = q * 4 + hi * 2;
      v2f a[4], b[2];
#pragma unroll
      for (int mt = 0; mt < 4; ++mt)
        a[mt] = *(const v2f*)&As[(wm0 + mt * 16 + l15) * LSTR + kbase];
#pragma unroll
      for (int nt = 0; nt < 2; ++nt)
        b[nt] = *(const v2f*)&Bs[(wn0 + nt * 16 + l15) * LSTR + kbase];
#pragma unroll
      for (int mt = 0; mt < 4; ++mt)
#pragma unroll
        for (int nt = 0; nt < 2; ++nt)
          acc[mt][nt] = __builtin_amdgcn_wmma_f32_16x16x4_f32(
              false, a[mt], false, b[nt], (short)0, acc[mt][nt], false, false);
    }
  }

  // Epilogue: bias per output COLUMN
#pragma unroll
  for (int mt = 0; mt < 4; ++mt)
#pragma unroll
    for (int nt = 0; nt < 2; ++nt) {
      int rg = Mblk + wm0 + mt * 16 + hi * 8;
      int cg = Nblk + wn0 + nt * 16 + l15;
      float bc = bias[cg];
#pragma unroll
      for (int r = 0; r < 8; ++r)
        Out[(size_t)(rg + r) * NMAT + cg] = acc[mt][nt][r] + bc;
    }
}

// ---------------- Greedy argmax masking ----------------
// mask[r,j] == (rowstep[r] > colstep[j]); single 1024-thread block with cached
// per-row maxima (matrix is L2-resident: 16MB << 192MB).
__global__ __launch_bounds__(1024) void greedy_kernel(
    const float* __restrict__ O, int* __restrict__ rowstep,
    int* __restrict__ colstep) {
  __shared__ float rowmax[NMAT];
  __shared__ int   rowarg[NMAT];
  __shared__ unsigned char coldead[NMAT];
  __shared__ float rv[1024];
  __shared__ int   ri[1024];
  __shared__ int   sel_j;
  const int tid = threadIdx.x;

  for (int r = tid; r < NMAT; r += 1024) {
    float m = NEG_INF; int a = 0;
    const float* row = O + (size_t)r * NMAT;
    for (int c = 0; c < NMAT; ++c) {
      float v = row[c];
      if (v > m) { m = v; a = c; }
    }
    rowmax[r] = m; rowarg[r] = a;
  }
  for (int c = tid; c < NMAT; c += 1024) coldead[c] = 0;
  __syncthreads();

  for (int k = 0; k < NMAT; ++k) {
    float bm = NEG_INF; int br = 0x7fffffff;
    for (int r = tid; r < NMAT; r += 1024) {
      float v = rowmax[r];
      if (v > bm || (v == bm && r < br)) { bm = v; br = r; }
    }
    rv[tid] = bm; ri[tid] = br;
    __syncthreads();
    for (int st = 512; st > 0; st >>= 1) {
      if (tid < st) {
        float v2 = rv[tid + st]; int r2 = ri[tid + st];
        if (v2 > rv[tid] || (v2 == rv[tid] && r2 < ri[tid])) { rv[tid] = v2; ri[tid] = r2; }
      }
      __syncthreads();
    }
    if (tid == 0) {
      int i = ri[0];
      int j = (i >= 0 && i < NMAT) ? rowarg[i] : -1;
      if (j >= 0) {
        rowstep[i] = k;
        colstep[j] = k;
        rowmax[i] = NEG_INF;
        coldead[j] = 1;
      }
      sel_j = j;
    }
    __syncthreads();
    const int j = sel_j;
    if (j >= 0) {
      for (int r = tid; r < NMAT; r += 1024) {
        if (rowmax[r] != NEG_INF && rowarg[r] == j) {
          float m = NEG_INF; int a = 0;
          const float* row = O + (size_t)r * NMAT;
          for (int c = 0; c < NMAT; ++c) {
            if (!coldead[c]) {
              float v = row[c];
              if (v > m) { m = v; a = c; }
            }
          }
          rowmax[r] = m; rowarg[r] = a;
        }
      }
    }
    __syncthreads();
  }
}

// ---------------- Masked softmax (in-place, one block per row) ----------------
__global__ __launch_bounds__(256) void masked_softmax_kernel(
    float* __restrict__ O, const int* __restrict__ rowstep,
    const int* __restrict__ colstep) {
  __shared__ float red[256];
  const int row = blockIdx.x;
  const int tid = threadIdx.x;
  const int sr  = rowstep[row];
  float* rowp = O + (size_t)row * NMAT;

  float m = NEG_INF;
  for (int c = tid; c < NMAT; c += 256)
    if (colstep[c] >= sr) m = fmaxf(m, rowp[c]);
  red[tid] = m;
  __syncthreads();
  for (int st = 128; st > 0; st >>= 1) {
    if (tid < st) red[tid] = fmaxf(red[tid], red[tid + st]);
    __syncthreads();
  }
  const float M = red[0];
  __syncthreads();

  float s = 0.f;
  for (int c = tid; c < NMAT; c += 256)
    if (colstep[c] >= sr) s += expf(rowp[c] - M);
  red[tid] = s;
  __syncthreads();
  for (int st = 128; st > 0; st >>= 1) {
    if (tid < st) red[tid] += red[tid + st];
    __syncthreads();
  }
  const float invS = 1.0f / red[0];
  __syncthreads();

  for (int c = tid; c < NMAT; c += 256) {
    float v = rowp[c];
    rowp[c] = (colstep[c] >= sr) ? expf(v - M) * invS : 0.0f;
  }
}

// ---------------- Launch ----------------

extern "C" void kernel_launch(void* const* d_in, const int* in_sizes, int n_in,
                              void* d_out, int out_size, void* d_ws, size_t ws_size,
                              hipStream_t stream) {
  (void)in_sizes; (void)n_in; (void)out_size; (void)ws_size;
  const size_t NN = (size_t)NMAT * NMAT;

  const float* x   = (const float*)d_in[0];
  const float* g1  = (const float*)d_in[1];
  const float* be1 = (const float*)d_in[2];
  const float* w1  = (const float*)d_in[3];
  const float* b1  = (const float*)d_in[4];
  const float* g2  = (const float*)d_in[5];
  const float* be2 = (const float*)d_in[6];
  const float* w2  = (const float*)d_in[7];
  const float* b2  = (const float*)d_in[8];

  float* out = (float*)d_out;
  float* ws  = (float*)d_ws;
  float* yb  = ws;                       // 2048x2048 intermediate Y
  float* sc1 = ws + NN;
  float* sh1 = sc1 + NMAT;
  float* sc2 = sh1 + NMAT;
  float* sh2 = sc2 + NMAT;
  int* rowstep = (int*)(sh2 + NMAT);
  int* colstep = rowstep + NMAT;

  dim3 ggrid(NMAT / BT, NMAT / BT);      // 16 x 16

  for (int l = 0; l < 2; ++l) {
    const float* src = (l == 0) ? x : out;   // activations ping through d_out
    colstats_kernel<<<NMAT / 256, 256, 0, stream>>>(src, g1 + l * NMAT, be1 + l * NMAT, sc1, sh1);
    gemm1_kernel<<<ggrid, 256, 0, stream>>>(w1 + (size_t)l * NN, src, sc1, sh1, b1 + l * NMAT, yb);
    rowstats_kernel<<<NMAT, 256, 0, stream>>>(yb, g2 + l * NMAT, be2 + l * NMAT, sc2, sh2);
    gemm2_kernel<<<ggrid, 256, 0, stream>>>(yb, sc2, sh2, w2 + (size_t)l * NN, b2 + l * NMAT, out);
  }

  greedy_kernel<<<1, 1024, 0, stream>>>(out, rowstep, colstep);
  masked_softmax_kernel<<<NMAT, 256, 0, stream>>>(out, rowstep, colstep);
}